// NP_Loss_1451698946788
// MI455X (gfx1250) — compile-verified
//
#include <hip/hip_runtime.h>
#include <hip/hip_bf16.h>
#include <math.h>

// ---------------------------------------------------------------------------
// NP loss for MI455X (gfx1250, wave32).
//   loss = -0.001*epoch*PSD(embeddings) + sum(-log(masked softmax of -D))/200
// Dominant compute: Gram = E @ E^T, E = 256 x 65536 fp32  -> fp32 WMMA 16x16x4.
// PSD: per-image 2D FFT (256-pt radix-2 in LDS), |F|^2 accumulated over batch.
// ---------------------------------------------------------------------------

#define BATCH  256
#define KDIM   65536           // 256*256 features per row
#define IMGN   256             // image side
#define KCHUNK 2048            // K elements per wave (32 chunks total)

typedef float v2f __attribute__((ext_vector_type(2)));
typedef float v8f __attribute__((ext_vector_type(8)));

// ------------------------------- init --------------------------------------
__global__ __launch_bounds__(256) void k_init(float* gram, float* psdacc, float* scal) {
    int i = blockIdx.x * blockDim.x + threadIdx.x;
    if (i < BATCH * BATCH)       gram[i]   = 0.0f;
    if (i < BATCH * BATCH)       psdacc[i] = 0.0f;
    if (i < 16)                  scal[i]   = 0.0f;
}

// --------------------------- Gram via WMMA ---------------------------------
// One wave -> 32x32 output tile (2x2 WMMA 16x16 accumulators) over a K-chunk.
// grid.x = 64 tiles (8x8), grid.y = K chunks, block = 32 (one wave, EXEC all 1s).
// Lane-dependent khalf is folded into the base pointers so the hot loop has a
// uniform, compile-time trip count (pure scalar loop control).
__global__ __launch_bounds__(32) void k_gram_wmma(const float* __restrict__ E,
                                                  float* __restrict__ G) {
    const int tile  = blockIdx.x;
    const int tm    = (tile >> 3) * 32;
    const int tn    = (tile & 7) * 32;
    const int lane  = threadIdx.x;        // 0..31
    const int r     = lane & 15;
    const int khalf = (lane >> 4) * 2;    // lanes 0-15: K+0,K+1 ; lanes 16-31: K+2,K+3
    const int kbase = blockIdx.y * KCHUNK + khalf;

    const float* A0 = E + (size_t)(tm +      r) * KDIM + kbase;
    const float* A1 = E + (size_t)(tm + 16 + r) * KDIM + kbase;
    const float* B0 = E + (size_t)(tn +      r) * KDIM + kbase;
    const float* B1 = E + (size_t)(tn + 16 + r) * KDIM + kbase;

    v8f c00 = {}; v8f c01 = {}; v8f c10 = {}; v8f c11 = {};

    #pragma unroll 4
    for (int it = 0; it < KCHUNK / 4; ++it) {
        v2f a0 = *(const v2f*)(A0 + 4 * it);   // {A[m][k], A[m][k+1]} per ISA 16x4 layout
        v2f a1 = *(const v2f*)(A1 + 4 * it);
        v2f b0 = *(const v2f*)(B0 + 4 * it);   // {B[k][n], B[k+1][n]} (B = E^T tile)
        v2f b1 = *(const v2f*)(B1 + 4 * it);
        c00 = __builtin_amdgcn_wmma_f32_16x16x4_f32(false, a0, false, b0, (short)0, c00, false, false);
        c01 = __builtin_amdgcn_wmma_f32_16x16x4_f32(false, a0, false, b1, (short)0, c01, false, false);
        c10 = __builtin_amdgcn_wmma_f32_16x16x4_f32(false, a1, false, b0, (short)0, c10, false, false);
        c11 = __builtin_amdgcn_wmma_f32_16x16x4_f32(false, a1, false, b1, (short)0, c11, false, false);
    }

    // C/D layout: VGPR v, lane L -> row = v + 8*(L>=16), col = L&15
    const int rowh = (lane >> 4) * 8;
    const int col  = lane & 15;
    for (int v = 0; v < 8; ++v) {
        int mr = rowh + v;
        atomicAdd(&G[(size_t)(tm +      mr) * BATCH + tn +      col], c00[v]);
        atomicAdd(&G[(size_t)(tm +      mr) * BATCH + tn + 16 + col], c01[v]);
        atomicAdd(&G[(size_t)(tm + 16 + mr) * BATCH + tn +      col], c10[v]);
        atomicAdd(&G[(size_t)(tm + 16 + mr) * BATCH + tn + 16 + col], c11[v]);
    }
}

// ----------------------------- 256-pt FFT ----------------------------------
__device__ __forceinline__ float2 cmul(float2 a, float2 b) {
    return make_float2(a.x * b.x - a.y * b.y, a.x * b.y + a.y * b.x);
}

// radix-2 DIT on 256 complex values in LDS; 128 threads, input bit-reversed.
__device__ __forceinline__ void fft256(float2* s, int t) {
    for (int len = 2; len <= 256; len <<= 1) {
        int half = len >> 1;
        int p = t & (half - 1);
        int a = ((t / half) * len) + p;
        int b = a + half;
        float ang = -6.283185307179586f * (float)p / (float)len;
        float2 w  = make_float2(cosf(ang), sinf(ang));
        float2 u  = s[a];
        float2 v  = cmul(w, s[b]);
        s[a] = make_float2(u.x + v.x, u.y + v.y);
        s[b] = make_float2(u.x - v.x, u.y - v.y);
        __syncthreads();
    }
}

// Row pass: FFT each row of each image in the chunk; write complex to W.
__global__ __launch_bounds__(128) void k_fft_rows(const float* __restrict__ X,
                                                  float2* __restrict__ W,
                                                  int imgBase) {
    __shared__ float2 s[256];
    const int t      = threadIdx.x;          // 0..127
    const int row    = blockIdx.x & 255;
    const int bLocal = blockIdx.x >> 8;
    const float* xr  = X + ((size_t)(imgBase + bLocal) * IMGN + row) * IMGN;
    const int i0 = __brev((unsigned)t)         >> 24;   // 8-bit bit reverse
    const int i1 = __brev((unsigned)(t + 128)) >> 24;
    s[t]       = make_float2(xr[i0], 0.0f);
    s[t + 128] = make_float2(xr[i1], 0.0f);
    __syncthreads();
    fft256(s, t);
    float2* wr = W + ((size_t)bLocal * IMGN + row) * IMGN;
    wr[t]       = s[t];
    wr[t + 128] = s[t + 128];
}

// Column pass: FFT each column, accumulate |F|^2 into the batch-sum PSD.
__global__ __launch_bounds__(128) void k_fft_cols(const float2* __restrict__ W,
                                                  float* __restrict__ psdacc) {
    __shared__ float2 s[256];
    const int t      = threadIdx.x;
    const int col    = blockIdx.x & 255;
    const int bLocal = blockIdx.x >> 8;
    const float2* wb = W + (size_t)bLocal * (IMGN * IMGN);
    const int i0 = __brev((unsigned)t)         >> 24;
    const int i1 = __brev((unsigned)(t + 128)) >> 24;
    s[t]       = wb[(size_t)i0 * IMGN + col];
    s[t + 128] = wb[(size_t)i1 * IMGN + col];
    __syncthreads();
    fft256(s, t);
    float2 f0 = s[t], f1 = s[t + 128];
    atomicAdd(&psdacc[(size_t)t * IMGN + col],         f0.x * f0.x + f0.y * f0.y);
    atomicAdd(&psdacc[(size_t)(t + 128) * IMGN + col], f1.x * f1.x + f1.y * f1.y);
}

// --------------------- PSD scalar: mean(log) - log(mean) --------------------
__global__ __launch_bounds__(256) void k_psd_reduce(const float* __restrict__ psdacc,
                                                    const int* __restrict__ epoch,
                                                    float* __restrict__ lossAcc) {
    __shared__ float slog[256];
    __shared__ float ssum[256];
    const int t = threadIdx.x;
    float al = 0.0f, as = 0.0f;
    for (int i = t; i < IMGN * IMGN; i += 256) {
        float v = psdacc[i] * (1.0f / 256.0f);   // mean over batch
        al += logf(v);
        as += v;
    }
    slog[t] = al; ssum[t] = as;
    __syncthreads();
    for (int s = 128; s > 0; s >>= 1) {
        if (t < s) { slog[t] += slog[t + s]; ssum[t] += ssum[t + s]; }
        __syncthreads();
    }
    if (t == 0) {
        const float k = (float)(IMGN * IMGN);
        float psd = slog[0] / k - logf(ssum[0] / k);
        psd = fminf(fmaxf(psd, 0.0f), 100.0f);
        atomicAdd(lossAcc, -0.001f * (float)epoch[0] * psd);
    }
}

// ----------------- off-diagonal distance softmax NLL term ------------------
__global__ __launch_bounds__(256) void k_row_softmax(const float* __restrict__ G,
                                                     const int* __restrict__ labels,
                                                     float* __restrict__ lossAcc) {
    __shared__ float red[256];
    const int i = blockIdx.x;
    const int j = threadIdx.x;
    const float gii = G[(size_t)i * BATCH + i];
    const float gjj = G[(size_t)j * BATCH + j];
    const float gij = G[(size_t)i * BATCH + j];
    float d2 = fmaxf(gii + gjj - 2.0f * gij, 0.0f);
    float D  = (d2 > 0.0f) ? sqrtf(d2) : 0.0f;
    const bool valid = (j != i);

    // stabilizer: max(-D) == -min(D) over valid entries
    red[j] = valid ? D : 3.4e38f;
    __syncthreads();
    for (int s = 128; s > 0; s >>= 1) { if (j < s) red[j] = fminf(red[j], red[j + s]); __syncthreads(); }
    const float mn = red[0];
    __syncthreads();

    const float e = valid ? expf(mn - D) : 0.0f;
    red[j] = e;
    __syncthreads();
    for (int s = 128; s > 0; s >>= 1) { if (j < s) red[j] += red[j + s]; __syncthreads(); }
    const float tot = red[0];
    __syncthreads();

    const int li = labels[i];
    red[j] = (valid && labels[j] == li) ? e : 0.0f;
    __syncthreads();
    for (int s = 128; s > 0; s >>= 1) { if (j < s) red[j] += red[j + s]; __syncthreads(); }
    if (j == 0) atomicAdd(lossAcc, -logf(red[0] / tot) * (1.0f / 200.0f));
}

__global__ void k_finalize(const float* lossAcc, float* out) {
    out[0] = lossAcc[0];
}

// ---------------------------------------------------------------------------
extern "C" void kernel_launch(void* const* d_in, const int* in_sizes, int n_in,
                              void* d_out, int out_size, void* d_ws, size_t ws_size,
                              hipStream_t stream) {
    const float* E      = (const float*)d_in[0];   // (256,1,256,256) fp32
    const int*   labels = (const int*)d_in[1];     // (256,) int
    const int*   epoch  = (const int*)d_in[2];     // scalar int
    float*       out    = (float*)d_out;

    char*   ws      = (char*)d_ws;
    float*  gram    = (float*)(ws);                       // 256 KB
    float*  psdacc  = (float*)(ws + (256u << 10));        // 256 KB
    float*  scal    = (float*)(ws + (512u << 10));        // scalars (lossAcc = scal[0])
    float2* W       = (float2*)(ws + (1u << 20));         // FFT staging, 512 KB / image

    // how many images of FFT staging fit in the remaining scratch
    size_t slotBytes = (size_t)IMGN * IMGN * sizeof(float2);
    long long avail  = (long long)ws_size - (1ll << 20);
    int nimg = (avail > 0) ? (int)(avail / (long long)slotBytes) : 1;
    if (nimg < 1)   nimg = 1;
    if (nimg > 256) nimg = 256;

    k_init<<<(BATCH * BATCH + 255) / 256, 256, 0, stream>>>(gram, psdacc, scal);

    // Gram: 64 tiles of 32x32, K split in 32 chunks of 2048 -> 2048 waves
    dim3 ggrid(64, KDIM / KCHUNK);
    k_gram_wmma<<<ggrid, 32, 0, stream>>>(E, gram);

    // 2D FFT + |F|^2 accumulation, chunked by available staging
    for (int base = 0; base < BATCH; base += nimg) {
        int cur = (BATCH - base < nimg) ? (BATCH - base) : nimg;
        k_fft_rows<<<cur * IMGN, 128, 0, stream>>>(E, W, base);
        k_fft_cols<<<cur * IMGN, 128, 0, stream>>>(W, psdacc);
    }

    k_psd_reduce<<<1, 256, 0, stream>>>(psdacc, epoch, scal);
    k_row_softmax<<<BATCH, 256, 0, stream>>>(gram, labels, scal);
    k_finalize<<<1, 1, 0, stream>>>(scal, out);
}